// QSEBlock_15401752724153
// MI455X (gfx1250) — compile-verified
//
#include <hip/hip_runtime.h>
#include <hip/hip_bf16.h>

typedef __attribute__((ext_vector_type(2))) float v2f;
typedef __attribute__((ext_vector_type(8))) float v8f;

#define CH   256
#define NQ   8
#define HW   3136           // 56*56
#define HW4  784            // HW/4
#define OUTC 16             // C/reduction

// ---------------------------------------------------------------------------
// Kernel 1: per-(b,c) mean over H*W.  One block per plane, float4 loads,
// deterministic LDS tree reduction.  grid = B*C = 16384 blocks.
// ---------------------------------------------------------------------------
__global__ void qse_mean_kernel(const float* __restrict__ x,
                                float* __restrict__ y) {
    __shared__ float sd[256];
    const int plane = blockIdx.x;                 // b*256 + c
    const int t = threadIdx.x;
    const float4* xp = (const float4*)(x + (size_t)plane * HW);
    float4 a = {0.f, 0.f, 0.f, 0.f};
    for (int i = t; i < HW4; i += 256) {
        float4 v = xp[i];
        a.x += v.x; a.y += v.y; a.z += v.z; a.w += v.w;
    }
    sd[t] = (a.x + a.y) + (a.z + a.w);
    __syncthreads();
    for (int s = 128; s > 0; s >>= 1) {
        if (t < s) sd[t] += sd[t + s];
        __syncthreads();
    }
    if (t == 0) y[plane] = sd[0] * (1.0f / (float)HW);
}

// ---------------------------------------------------------------------------
// Kernel 2: per-batch squeeze math.  One block (256 threads) per batch.
// Thread c replays the CNOT ladder on its own basis index (pure integer ops),
// giving p = P(c); ez[i] = sum_c (y[c]^2 / sumsq) * (1 - 2*bit_{7-i}(p)).
// Then qr[b][k] = relu(ez[k % 8]) for k = 0..15.
// ---------------------------------------------------------------------------
__global__ void qse_squeeze_kernel(const float* __restrict__ y,
                                   float* __restrict__ qr) {
    __shared__ float sd[256];
    __shared__ float ez[NQ];
    const int b = blockIdx.x;
    const int c = threadIdx.x;

    const float yc = y[b * CH + c];
    const float yc2 = yc * yc;
    sd[c] = yc2;
    __syncthreads();
    for (int s = 128; s > 0; s >>= 1) {
        if (c < s) sd[c] += sd[c + s];
        __syncthreads();
    }
    const float sumsq = sd[0];
    __syncthreads();
    const float v = yc2 / sumsq;                  // amp[c]^2

    // Replay circuit: for i<j, CNOT(i,j) then CNOT(j,i); wire 0 = MSB.
    int p = c;
    #pragma unroll
    for (int i = 0; i < NQ; ++i)
        #pragma unroll
        for (int j = i + 1; j < NQ; ++j) {
            if ((p >> (NQ - 1 - i)) & 1) p ^= 1 << (NQ - 1 - j);
            if ((p >> (NQ - 1 - j)) & 1) p ^= 1 << (NQ - 1 - i);
        }

    for (int i = 0; i < NQ; ++i) {
        const float sgn = ((p >> (NQ - 1 - i)) & 1) ? -1.0f : 1.0f;
        sd[c] = v * sgn;
        __syncthreads();
        for (int s = 128; s > 0; s >>= 1) {
            if (c < s) sd[c] += sd[c + s];
            __syncthreads();
        }
        if (c == 0) ez[i] = sd[0];
        __syncthreads();
    }
    if (c < OUTC) qr[b * OUTC + c] = fmaxf(ez[c & (NQ - 1)], 0.0f);
}

// ---------------------------------------------------------------------------
// Kernel 3: excitation GEMM via WMMA.
//   s[b][c] = sigmoid( sum_k qr[b][k] * W[c][k] ),  64x16x256 f32 GEMM.
// One wave per 16x16 output tile; K=16 as 4 chained V_WMMA_F32_16X16X4_F32.
// Layouts (ISA 7.12.2, wave32):
//   A 16x4 f32 : lane l (half=l/16, lm=l%16): vgpr r = A[M=lm][K=r+2*half]
//   B 4x16 f32 : lane l: vgpr r = B[K=r+2*half][N=lm]
//   D 16x16 f32: vgpr v = D[M=v+8*half][N=lm]
// grid = 16 blocks x 128 threads (4 waves) = 64 tiles.
// ---------------------------------------------------------------------------
__global__ void qse_excite_wmma_kernel(const float* __restrict__ qr,
                                       const float* __restrict__ Wm,
                                       float* __restrict__ s_out) {
    const int lane = threadIdx.x & 31;
    const int wave = threadIdx.x >> 5;            // 0..3
    const int tile = blockIdx.x * 4 + wave;       // 0..63
    const int bt = tile >> 4;                     // batch tile  0..3
    const int ct = tile & 15;                     // channel tile 0..15
    const int half = lane >> 4;
    const int lm = lane & 15;

    v8f acc = {};
    #pragma unroll
    for (int kk = 0; kk < 4; ++kk) {
        const int kb = kk * 4 + 2 * half;
        v2f a, bmat;
        a.x    = qr[(bt * 16 + lm) * OUTC + kb + 0];
        a.y    = qr[(bt * 16 + lm) * OUTC + kb + 1];
        bmat.x = Wm[(ct * 16 + lm) * OUTC + kb + 0];   // B[k][n] = W[n][k]
        bmat.y = Wm[(ct * 16 + lm) * OUTC + kb + 1];
        acc = __builtin_amdgcn_wmma_f32_16x16x4_f32(
            /*neg_a=*/false, a, /*neg_b=*/false, bmat,
            /*c_mod=*/(short)0, acc, /*reuse_a=*/false, /*reuse_b=*/false);
    }

    #pragma unroll
    for (int v = 0; v < 8; ++v) {
        const int row = bt * 16 + v + 8 * half;   // batch index
        const int col = ct * 16 + lm;             // channel index
        s_out[row * CH + col] = 1.0f / (1.0f + expf(-acc[v]));
    }
}

// ---------------------------------------------------------------------------
// Kernel 4: out = x * s[b][c] broadcast.  Pure streaming, float4.
// grid = B*C*HW4 / 256 = 50176 blocks (exact).
// ---------------------------------------------------------------------------
__global__ void qse_scale_kernel(const float* __restrict__ x,
                                 const float* __restrict__ s,
                                 float* __restrict__ out) {
    const size_t f = (size_t)blockIdx.x * blockDim.x + threadIdx.x; // float4 idx
    const int plane = (int)(f / HW4);             // b*256 + c
    const float sc = s[plane];
    float4 v = ((const float4*)x)[f];
    v.x *= sc; v.y *= sc; v.z *= sc; v.w *= sc;
    ((float4*)out)[f] = v;
}

// ---------------------------------------------------------------------------
extern "C" void kernel_launch(void* const* d_in, const int* in_sizes, int n_in,
                              void* d_out, int out_size, void* d_ws, size_t ws_size,
                              hipStream_t stream) {
    const float* x  = (const float*)d_in[0];      // 64*256*56*56
    const float* Wm = (const float*)d_in[1];      // 256*16
    float* out = (float*)d_out;

    const int B = 64;
    float* y_ws  = (float*)d_ws;                  // B*CH       = 16384 floats
    float* qr_ws = y_ws + B * CH;                 // B*OUTC     = 1024 floats
    float* s_ws  = qr_ws + B * OUTC;              // B*CH       = 16384 floats

    qse_mean_kernel<<<B * CH, 256, 0, stream>>>(x, y_ws);
    qse_squeeze_kernel<<<B, 256, 0, stream>>>(y_ws, qr_ws);
    qse_excite_wmma_kernel<<<16, 128, 0, stream>>>(qr_ws, Wm, s_ws);

    const size_t nf4 = (size_t)B * CH * HW4;      // 12,845,056 float4s
    qse_scale_kernel<<<(unsigned)(nf4 / 256), 256, 0, stream>>>(x, s_ws, out);
}